// NTN_52475910423088
// MI455X (gfx1250) — compile-verified
//
#include <hip/hip_runtime.h>

// NTN fused kernel for MI455X (gfx1250), compile-only target.
//
// h_k = x1^T W[:,:,k] x2  (W is [1024,1024,256] fp32, 1 GiB -> HBM bound, 0.5 FLOP/B)
// out = sum_k U_k * relu(h_k + (V @ [x1;x2])_k + b_k)
//
// Kernel 1: block b owns slab W[b,:,:]; 16 waves each own a 16-wide k tile.
//   WMMA f32 16x16x4: B = 4 W-rows x 16 k-cols, A = x2[j0..j0+3] broadcast to
//   all 16 M rows => every accumulator row equals partial h[k0+n]. Scale by
//   x1[b] at the end, write partials[b][k] to workspace (deterministic, no atomics).
// Kernel 2: reduce partials over b, add V-GEMV + bias, relu, dot with U.

#define DD 1024
#define KK 256

typedef __attribute__((ext_vector_type(2))) float v2f;
typedef __attribute__((ext_vector_type(8))) float v8f;

__global__ __launch_bounds__(512) void ntn_bilinear_wmma(
    const float* __restrict__ x1, const float* __restrict__ x2,
    const float* __restrict__ W, float* __restrict__ partials)
{
    __shared__ float s_x2[DD];
    const int tid = threadIdx.x;
    // stage x2 (4 KB) into LDS: 512 threads * 2 elements
    s_x2[tid]        = x2[tid];
    s_x2[tid + 512]  = x2[tid + 512];
    __syncthreads();

    const int b    = blockIdx.x;       // i index (slab W[b,:,:])
    const int wave = tid >> 5;         // 0..15 -> k tile
    const int lane = tid & 31;
    const int k0   = wave * 16;
    const int n    = lane & 15;        // N column within tile
    const int rsel = (lane >> 4) << 1; // 0 for lanes 0-15, 2 for lanes 16-31
                                       // (A/B VGPR0 holds K=0|2, VGPR1 holds K=1|3)

    const float* Wb = W + (size_t)b * DD * KK;

    v8f acc = {0.f, 0.f, 0.f, 0.f, 0.f, 0.f, 0.f, 0.f};

    // per-lane pointer to W[b, j0+rsel, k0+n]; second element one row (+KK) below
    const float* bp = Wb + (size_t)rsel * KK + (k0 + n);

    for (int j0 = 0; j0 < DD; j0 += 4) {
        // A operand: A[m, r] = x2[j0 + r] for every m (rows replicated)
        v2f a;
        a.x = s_x2[j0 + rsel];          // K = 0 | 2
        a.y = s_x2[j0 + rsel + 1];      // K = 1 | 3
        // B operand: B[r, n] = W[b, j0 + r, k0 + n]
        v2f bm;
        bm.x = bp[0];                   // row j0 + rsel
        bm.y = bp[KK];                  // row j0 + rsel + 1
        // prefetch the W stream ~8 rows (2 iterations) ahead
        __builtin_prefetch(bp + 8 * KK, 0, 0);
        // D = A x B + C ; 16x16 f32 accumulator, all rows identical
        acc = __builtin_amdgcn_wmma_f32_16x16x4_f32(
            /*neg_a=*/false, a, /*neg_b=*/false, bm,
            /*c_mod=*/(short)0, acc, /*reuse_a=*/false, /*reuse_b=*/false);
        bp += 4 * KK;
    }

    // Every accumulator row holds the same partial; take M=0 row from lanes 0-15.
    if (lane < 16) {
        partials[(size_t)b * KK + (k0 + n)] = x1[b] * acc[0];
    }
}

__global__ __launch_bounds__(256) void ntn_finish(
    const float* __restrict__ x1, const float* __restrict__ x2,
    const float* __restrict__ V, const float* __restrict__ U,
    const float* __restrict__ bias, const float* __restrict__ partials,
    float* __restrict__ out)
{
    const int k = threadIdx.x;  // 0..255

    // h_k = sum over 1024 block partials
    float h = 0.f;
    for (int bb = 0; bb < DD; ++bb)
        h += partials[(size_t)bb * KK + k];

    // v_k = V[k,:] . concat(x1, x2)   (row is 8 KB, per-thread sequential)
    float v = 0.f;
    const float* Vr = V + (size_t)k * (2 * DD);
    for (int t = 0; t < DD; ++t) v = fmaf(Vr[t],      x1[t], v);
    for (int t = 0; t < DD; ++t) v = fmaf(Vr[DD + t], x2[t], v);

    float m = h + v + bias[k];
    float f = m > 0.f ? m : 0.f;

    __shared__ float red[KK];
    red[k] = U[k] * f;
    __syncthreads();
    for (int s = KK / 2; s > 0; s >>= 1) {
        if (k < s) red[k] += red[k + s];
        __syncthreads();
    }
    if (k == 0) out[0] = red[0];
}

extern "C" void kernel_launch(void* const* d_in, const int* in_sizes, int n_in,
                              void* d_out, int out_size, void* d_ws, size_t ws_size,
                              hipStream_t stream) {
    (void)in_sizes; (void)n_in; (void)out_size; (void)ws_size;
    const float* x1 = (const float*)d_in[0];
    const float* x2 = (const float*)d_in[1];
    const float* W  = (const float*)d_in[2];
    const float* V  = (const float*)d_in[3];
    const float* U  = (const float*)d_in[4];
    const float* b  = (const float*)d_in[5];
    float* partials = (float*)d_ws;         // 1024 * 256 * 4 B = 1 MiB scratch
    float* out      = (float*)d_out;

    ntn_bilinear_wmma<<<DD, 512, 0, stream>>>(x1, x2, W, partials);
    ntn_finish<<<1, KK, 0, stream>>>(x1, x2, V, U, b, partials, out);
}